// EA_embedding_patches_57028575756391
// MI455X (gfx1250) — compile-verified
//
#include <hip/hip_runtime.h>

typedef __attribute__((ext_vector_type(2))) float v2f;
typedef __attribute__((ext_vector_type(8))) float v8f;

#define C_DIM 2048
#define B_DIM 16
#define P_DIM 16
#define CR    128
#define HW    64

// ---------------------------------------------------------------------------
// Phase A: y[b][c] = mean of x[b,c,16:32,16:32].  One wave32 per (b,c) pair.
// lane -> col = 16 + (lane&15), rows 16 + (lane>>4) + 2*i, i = 0..7
// ---------------------------------------------------------------------------
__global__ void patch_mean_kernel(const float* __restrict__ x,
                                  float* __restrict__ y) {
  const int tid  = threadIdx.x;
  const int wave = tid >> 5;
  const int lane = tid & 31;
  const int bc   = blockIdx.x * 8 + wave;          // 0 .. B*C-1 = 32767
  const int col  = 16 + (lane & 15);
  const int row0 = 16 + (lane >> 4);
  const float* base = x + (size_t)bc * (HW * HW);
  float s = 0.0f;
#pragma unroll
  for (int i = 0; i < 8; ++i)
    s += base[(row0 + 2 * i) * HW + col];
#pragma unroll
  for (int off = 16; off >= 1; off >>= 1)
    s += __shfl_xor(s, off, 32);
  if (lane == 0) y[bc] = s * (1.0f / 256.0f);
}

// ---------------------------------------------------------------------------
// Phase B: h[p][b][k] = relu( sum_c W1[p][k][c] * y[b][c] )
// D(16b x 16k) = A(y: 16b x 2048c) * B(W1^T: 2048c x 16k)
// One block per (p, ktile); 8 waves split C, LDS-reduce partials.
// f32 WMMA frag layout (A and B): lane l, VGPR j holds K = 2*(l>>4)+j,
// M/N = l&15  -> both fragments are contiguous float2 loads here.
// ---------------------------------------------------------------------------
__global__ void gemm1_relu_kernel(const float* __restrict__ W1,
                                  const float* __restrict__ y,
                                  float* __restrict__ h) {
  __shared__ float red[8 * 256];
  const int tid   = threadIdx.x;
  const int wave  = tid >> 5;
  const int lane  = tid & 31;
  const int p     = blockIdx.x >> 3;     // 0..15
  const int ktile = blockIdx.x & 7;      // 0..7
  const int k0    = ktile * 16;
  const int m     = lane & 15;           // M (=b) for A, N (=k offset) for B
  const int half  = lane >> 4;           // K-pair selector

  const float* ybase  = y  + (size_t)m * C_DIM + 2 * half;                // A: b = m
  const float* w1base = W1 + ((size_t)(p * CR + k0 + m)) * C_DIM + 2 * half; // B: k col

  v8f acc = {};
  const int cstart = wave * 256;         // each wave covers 256 of C=2048
  for (int cc = 0; cc < 64; ++cc) {
    const int c0 = cstart + cc * 4;
    v2f a = *(const v2f*)(ybase + c0);
    v2f b = *(const v2f*)(w1base + c0);
    acc = __builtin_amdgcn_wmma_f32_16x16x4_f32(false, a, false, b,
                                                (short)0, acc, false, false);
  }

  // cross-wave reduction of the 8 partial C tiles
#pragma unroll
  for (int i = 0; i < 8; ++i) red[wave * 256 + lane * 8 + i] = acc[i];
  __syncthreads();

  const int i = tid >> 5;                // accumulator slot 0..7
  const int l = tid & 31;                // lane of the tile
  float s = 0.0f;
#pragma unroll
  for (int w = 0; w < 8; ++w) s += red[w * 256 + l * 8 + i];

  const int b_out = i + 8 * (l >> 4);    // D layout: VGPR i -> M=i / i+8
  const int k_out = k0 + (l & 15);       // N = lane
  h[((size_t)(p * B_DIM + b_out)) * CR + k_out] = fmaxf(s, 0.0f);
}

// ---------------------------------------------------------------------------
// Phase C: att[p][b][c] = sigmoid( sum_k W2[p][c][k] * h[p][b][k] )
// D(16b x 16c) = A(h: 16b x 128k) * B(W2^T: 128k x 16c)
// One wave per (p, ctile); K=128 -> 32 WMMA steps. Coalesced stores in c.
// ---------------------------------------------------------------------------
__global__ void gemm2_sigmoid_kernel(const float* __restrict__ W2,
                                     const float* __restrict__ h,
                                     float* __restrict__ out) {
  const int tid  = threadIdx.x;
  const int wave = tid >> 5;
  const int lane = tid & 31;
  const int tile = blockIdx.x * 8 + wave;   // 0..2047
  const int p    = tile >> 7;               // 0..15
  const int c0   = (tile & 127) * 16;       // c tile origin
  const int m    = lane & 15;
  const int half = lane >> 4;

  const float* hbase  = h  + ((size_t)(p * B_DIM + m)) * CR + 2 * half;       // A: b = m
  const float* w2base = W2 + ((size_t)(p * C_DIM + c0 + m)) * CR + 2 * half;  // B: c = c0+m

  v8f acc = {};
#pragma unroll
  for (int kk = 0; kk < 32; ++kk) {
    v2f a = *(const v2f*)(hbase + kk * 4);
    v2f b = *(const v2f*)(w2base + kk * 4);
    acc = __builtin_amdgcn_wmma_f32_16x16x4_f32(false, a, false, b,
                                                (short)0, acc, false, false);
  }

  // D layout: slot i, lane l -> b = i + 8*(l>>4), c = c0 + (l&15)
#pragma unroll
  for (int i = 0; i < 8; ++i) {
    const int b_out = i + 8 * half;
    const float v = 1.0f / (1.0f + __expf(-acc[i]));
    out[((size_t)(p * B_DIM + b_out)) * C_DIM + c0 + m] = v;
  }
}

// ---------------------------------------------------------------------------
extern "C" void kernel_launch(void* const* d_in, const int* in_sizes, int n_in,
                              void* d_out, int out_size, void* d_ws, size_t ws_size,
                              hipStream_t stream) {
  (void)in_sizes; (void)n_in; (void)out_size; (void)ws_size;
  const float* x  = (const float*)d_in[0];   // (16, 2048, 64, 64)
  const float* W1 = (const float*)d_in[1];   // (16, 128, 2048)
  const float* W2 = (const float*)d_in[2];   // (16, 2048, 128)
  float* out = (float*)d_out;                // (16, 16, 2048, 1, 1)

  float* y = (float*)d_ws;                   // 16*2048   floats (128 KB)
  float* h = (float*)d_ws + B_DIM * C_DIM;   // 16*16*128 floats (128 KB)

  // B*C = 32768 patches, 8 waves/block -> 4096 blocks
  patch_mean_kernel<<<4096, 256, 0, stream>>>(x, y);
  // 16 p * 8 ktiles = 128 blocks, 8 waves split K(=C) per block
  gemm1_relu_kernel<<<128, 256, 0, stream>>>(W1, y, h);
  // 16 p * 128 ctiles = 2048 wave-tiles -> 256 blocks of 8 waves
  gemm2_sigmoid_kernel<<<256, 256, 0, stream>>>(W2, h, out);
}